// MinibatchDiscrimination_3401614099036
// MI455X (gfx1250) — compile-verified
//
#include <hip/hip_runtime.h>

typedef float v2f __attribute__((ext_vector_type(2)));
typedef float v8f __attribute__((ext_vector_type(8)));

#define N_  512
#define D_  512
#define O_  64
#define H_  16
#define OH_ 1024   // O_ * H_

// -----------------------------------------------------------------------------
// Kernel 1: M[N_, OH_] = x[N_, D_] @ W[D_, OH_]   (W = T viewed as [D][O*H])
// Uses V_WMMA_F32_16X16X4_F32 (CDNA5 fp32 WMMA, wave32).
// grid = (OH_/64, N_/16), block = 128 threads = 4 waves.
// Wave w computes the 16x16 tile at rows [by*16, +16), cols [bx*64 + w*16, +16).
// A-panel (16 x 512 f32 = 32KB) staged in LDS via GLOBAL_LOAD_ASYNC_TO_LDS_B128
// (ASYNCcnt-tracked DMA into LDS, no VGPR round-trip), shared by all 4 waves.
//
// Per-lane WMMA operand layout (ISA 7.12.2, wave32):
//   A 16x4 (2 VGPRs): lanes 0-15 -> row M=lane, K={0,1}; lanes 16-31 -> K={2,3}
//   B 4x16 (2 VGPRs): lanes 0-15 -> col N=lane, K={0,1}; lanes 16-31 -> K={2,3}
//   C/D 16x16 (8 VGPRs): vgpr r -> M=r (lanes 0-15), M=r+8 (lanes 16-31), N=lane%16
// -----------------------------------------------------------------------------
__global__ __launch_bounds__(128) void gemm_wmma_f32(const float* __restrict__ x,
                                                     const float* __restrict__ W,
                                                     float* __restrict__ M) {
  __shared__ float As[16 * D_];  // 32 KB, at LDS offset 0

  const int t    = threadIdx.x;
  const int row0 = blockIdx.y * 16;

  // Async A-panel stage: 16 rows x 512 K = 2048 x b128, 16 per thread.
  // GV mode: per-lane 64-bit global address, per-lane LDS byte offset in VGPR;
  // hardware adds LDS_BASE (As is the only shared object -> offset 0).
  {
    const float4* __restrict__ xsrc = (const float4*)(x + (size_t)row0 * D_);
#pragma unroll
    for (int i = 0; i < 16; ++i) {
      const float4* g = xsrc + t + i * 128;
      unsigned ldsoff = (unsigned)((t + i * 128) * 16);
      asm volatile("global_load_async_to_lds_b128 %0, %1, off"
                   :
                   : "v"(ldsoff), "v"(g)
                   : "memory");
    }
    // Escape As's address into the asm: forces LLVM to assume the async DMA
    // writes the shared array (otherwise the never-stored LDS array is folded
    // to poison and deleted, as seen in round 2's group_segment_fixed_size=0).
    asm volatile("s_wait_asynccnt 0" : : "v"(As) : "memory");
  }
  __syncthreads();

  const int wave = t >> 5;
  const int lane = t & 31;
  const int half = lane >> 4;   // 0: K pair {0,1}; 1: K pair {2,3}
  const int l    = lane & 15;
  const int col  = blockIdx.x * 64 + wave * 16 + l;

  const float* __restrict__ arow = As + l * D_ + 2 * half;              // A: row l, +K offset
  const float* __restrict__ bcol = W + col + (size_t)(2 * half) * OH_;  // B: col, +K offset

  v8f acc = {};
#pragma unroll 4
  for (int k = 0; k < D_; k += 4) {
    // Prefetch B ~32 K-rows (128 KB) ahead, once per 16 K.
    if ((k & 15) == 0 && (k + 32) < D_) {
      __builtin_prefetch(bcol + (size_t)(k + 32) * OH_, 0, 1);
    }
    v2f a, b;
    a.x = arow[k];
    a.y = arow[k + 1];
    b.x = bcol[(size_t)k * OH_];
    b.y = bcol[(size_t)(k + 1) * OH_];
    // D = A(16x4) * B(4x16) + C ; 8 args: neg_a, A, neg_b, B, c_mod, C, reuse_a, reuse_b
    acc = __builtin_amdgcn_wmma_f32_16x16x4_f32(false, a, false, b, (short)0, acc,
                                                false, false);
  }

  // Scatter the 16x16 f32 accumulator tile.
  float* __restrict__ mcol = M + col;
#pragma unroll
  for (int r = 0; r < 8; ++r) {
    mcol[(size_t)(row0 + r + 8 * half) * OH_] = acc[r];
  }
}

// -----------------------------------------------------------------------------
// Kernel 2: out[i, D_ + o] = sum_j exp(-sum_h |M[i,o,h] - M[j,o,h]|) - 1
// grid = O_ blocks (one per o), block = 512 threads (one per i, 16 waves).
// The full slice M[:, o, :] (512 x 16 f32 = 32 KB) is cached in LDS; the inner
// j-loop reads it with wave-uniform addresses (broadcast, conflict-free).
// Staging stays synchronous here: each thread needs its own 16 values in
// registers anyway, so the VGPR round-trip is free.
// -----------------------------------------------------------------------------
__global__ __launch_bounds__(512) void pairwise_exp_l1(const float* __restrict__ M,
                                                       float* __restrict__ out) {
  __shared__ float Ms[N_ * H_];  // 32 KB

  const int o = blockIdx.x;
  const int i = threadIdx.x;

  const float4* __restrict__ msrc = (const float4*)(M + (size_t)i * OH_ + o * H_);
  float4 v0 = msrc[0];
  float4 v1 = msrc[1];
  float4 v2 = msrc[2];
  float4 v3 = msrc[3];

  float4* __restrict__ mlds = (float4*)Ms;
  mlds[i * 4 + 0] = v0;
  mlds[i * 4 + 1] = v1;
  mlds[i * 4 + 2] = v2;
  mlds[i * 4 + 3] = v3;

  float mi[16];
  mi[0] = v0.x; mi[1] = v0.y; mi[2]  = v0.z; mi[3]  = v0.w;
  mi[4] = v1.x; mi[5] = v1.y; mi[6]  = v1.z; mi[7]  = v1.w;
  mi[8] = v2.x; mi[9] = v2.y; mi[10] = v2.z; mi[11] = v2.w;
  mi[12] = v3.x; mi[13] = v3.y; mi[14] = v3.z; mi[15] = v3.w;

  __syncthreads();

  float s = 0.0f;
#pragma unroll 2
  for (int j = 0; j < N_; ++j) {
    const float* __restrict__ mj = Ms + j * H_;
    float l1 = 0.0f;
#pragma unroll
    for (int h = 0; h < H_; ++h) {
      l1 += __builtin_fabsf(mi[h] - mj[h]);
    }
    s += __expf(-l1);  // v_exp_f32 path
  }

  // self term exp(0)=1 is included in s; reference subtracts it.
  out[(size_t)i * (D_ + O_) + D_ + o] = s - 1.0f;
}

// -----------------------------------------------------------------------------
// Kernel 3: out[:, 0:D_] = x   (float4 copy; out row stride = D_+O_ = 576)
// -----------------------------------------------------------------------------
__global__ __launch_bounds__(256) void copy_x(const float* __restrict__ x,
                                              float* __restrict__ out) {
  const int idx = blockIdx.x * 256 + threadIdx.x;  // N_*D_/4 = 65536 total
  const int row = idx >> 7;        // / (D_/4)
  const int c   = (idx & 127) << 2;
  *(float4*)(out + (size_t)row * (D_ + O_) + c) =
      *(const float4*)(x + (size_t)row * D_ + c);
}

extern "C" void kernel_launch(void* const* d_in, const int* in_sizes, int n_in,
                              void* d_out, int out_size, void* d_ws, size_t ws_size,
                              hipStream_t stream) {
  const float* x = (const float*)d_in[0];        // [N_, D_]
  const float* T = (const float*)d_in[1];        // [D_, O_, H_] == W[D_][OH_]
  float* out = (float*)d_out;                    // [N_, D_+O_]
  float* M = (float*)d_ws;                       // needs N_*OH_*4 = 2 MB scratch

  dim3 g1(OH_ / 64, N_ / 16);                    // 16 x 32 = 512 blocks
  gemm_wmma_f32<<<g1, 128, 0, stream>>>(x, T, M);

  pairwise_exp_l1<<<O_, 512, 0, stream>>>(M, out);

  copy_x<<<(N_ * D_ / 4) / 256, 256, 0, stream>>>(x, out);
}